// ModulatedDeformConv_10780367912993
// MI455X (gfx1250) — compile-verified
//
#include <hip/hip_runtime.h>
#include <stdint.h>

#define BATCH  8
#define CHN    128
#define HWDIM  64
#define KSZ    3
#define KK     9
#define PADSZ  1
#define ROWS   4
#define TPX    (ROWS * HWDIM)   // 256 threads = pixels per block
#define CSPLIT 4
#define CPB    (CHN / CSPLIT)   // 32 channels per block

typedef unsigned int u32;
typedef u32 v4u __attribute__((ext_vector_type(4)));
typedef int  v4i __attribute__((ext_vector_type(4)));
typedef int  v8i __attribute__((ext_vector_type(8)));

#if defined(__AMDGCN__) && __has_builtin(__builtin_amdgcn_tensor_load_to_lds)
#define HAVE_TDM 1
#else
#define HAVE_TDM 0
#endif

__device__ __forceinline__ float ldg_boff(const float* p, int boff) {
  return *(const float*)((const char*)p + boff);
}

#if HAVE_TDM
// 2-D TDM descriptor load: tensor [tdim1, tdim0] (row stride stride0, f32),
// tile [tile1, tile0] starting at gptr, packed contiguously at LDS lds_addr.
__device__ __forceinline__ void tdm_load_2d_f32(u32 lds_addr, const void* gptr,
                                                u32 tdim0, u32 tdim1,
                                                u32 tile0, u32 tile1,
                                                u32 stride0) {
  uint64_t ga = (uint64_t)(uintptr_t)gptr;
  v4u g0;
  g0[0] = 1u;                                             // count=1, user D#
  g0[1] = lds_addr;                                       // LDS byte address
  g0[2] = (u32)ga;                                        // global_addr[31:0]
  g0[3] = (u32)((ga >> 32) & 0x01FFFFFFu) | (2u << 30);   // [56:32] | type=2
  v8i g1;
  g1[0] = (int)(2u << 16);                                // data_size=4B, wg_mask=0
  g1[1] = (int)((tdim0 & 0xFFFFu) << 16);                 // tensor_dim0 lo
  g1[2] = (int)((tdim0 >> 16) | ((tdim1 & 0xFFFFu) << 16));
  g1[3] = (int)((tdim1 >> 16) | (tile0 << 16));           // tile_dim0
  g1[4] = (int)(tile1 & 0xFFFFu);                         // tile_dim1, tile_dim2=0
  g1[5] = (int)stride0;                                   // tensor_dim0_stride lo
  g1[6] = 0;
  g1[7] = 0;
  v4i z4 = {0, 0, 0, 0};
#if __clang_major__ >= 23
  v8i z8 = {0, 0, 0, 0, 0, 0, 0, 0};
  __builtin_amdgcn_tensor_load_to_lds(g0, g1, z4, z4, z8, 0);
#else
  __builtin_amdgcn_tensor_load_to_lds(g0, g1, z4, z4, 0);
#endif
}
#endif

__global__ __launch_bounds__(TPX) void ModulatedDeformConv_10780367912993_kernel(
    const float* __restrict__ x, const float* __restrict__ off,
    const float* __restrict__ msk, const float* __restrict__ wgt,
    float* __restrict__ out) {
  __shared__ float s_off[2 * KK * TPX];   // [18][256] staged offsets
  __shared__ float s_msk[KK * TPX];       // [9][256] staged masks

  const int b    = blockIdx.x;
  const int row0 = blockIdx.y * ROWS;
  const int c0   = blockIdx.z * CPB;
  const int t    = threadIdx.x;
  const size_t plane = (size_t)HWDIM * HWDIM;   // 4096

#if HAVE_TDM
  if (t < 32) {   // one wave issues both TDM DMAs, waits on TENSORcnt
    tdm_load_2d_f32((u32)(uintptr_t)s_off,
                    off + (size_t)b * 2 * KK * plane + (size_t)row0 * HWDIM,
                    (u32)plane, 2 * KK, TPX, 2 * KK, (u32)plane);
    tdm_load_2d_f32((u32)(uintptr_t)s_msk,
                    msk + (size_t)b * KK * plane + (size_t)row0 * HWDIM,
                    (u32)plane, KK, TPX, KK, (u32)plane);
    __builtin_amdgcn_s_wait_tensorcnt(0);
  }
#else
  for (int i = t; i < 2 * KK * TPX; i += TPX) {
    int ch = i / TPX, p = i - ch * TPX;
    s_off[i] = off[((size_t)b * 2 * KK + ch) * plane + (size_t)row0 * HWDIM + p];
  }
  for (int i = t; i < KK * TPX; i += TPX) {
    int ch = i / TPX, p = i - ch * TPX;
    s_msk[i] = msk[((size_t)b * KK + ch) * plane + (size_t)row0 * HWDIM + p];
  }
#endif
  __syncthreads();

  const int row = row0 + (t >> 6);
  const int col = t & (HWDIM - 1);

  // Per-pixel sampling metadata, computed once, reused for all 32 channels.
  int   boff[KK][4];   // corner byte offsets within one x-plane
  float coef[KK][4];   // mask * bilinear * validity
#pragma unroll
  for (int k = 0; k < KK; ++k) {
    const float py  = (float)(row - PADSZ + k / KSZ) + s_off[(2 * k) * TPX + t];
    const float px  = (float)(col - PADSZ + k % KSZ) + s_off[(2 * k + 1) * TPX + t];
    const float m   = s_msk[k * TPX + t];
    const float y0f = floorf(py);
    const float x0f = floorf(px);
    const float dy  = py - y0f;
    const float dx  = px - x0f;
    const int   y0  = (int)y0f;
    const int   x0  = (int)x0f;
    const float wy[2] = {1.0f - dy, dy};
    const float wx[2] = {1.0f - dx, dx};
#pragma unroll
    for (int cy = 0; cy < 2; ++cy) {
#pragma unroll
      for (int cx = 0; cx < 2; ++cx) {
        const int  yy = y0 + cy;
        const int  xx = x0 + cx;
        const bool v  = (yy >= 0) && (yy < HWDIM) && (xx >= 0) && (xx < HWDIM);
        const int  yc = yy < 0 ? 0 : (yy > HWDIM - 1 ? HWDIM - 1 : yy);
        const int  xc = xx < 0 ? 0 : (xx > HWDIM - 1 ? HWDIM - 1 : xx);
        boff[k][cy * 2 + cx] = (yc * HWDIM + xc) * 4;
        coef[k][cy * 2 + cx] = v ? m * wy[cy] * wx[cx] : 0.0f;
      }
    }
  }

  const float* xb = x   + ((size_t)b * CHN + c0) * plane;   // uniform saddr base
  const float* wb = wgt + ((size_t)b * CHN + c0) * KK;      // block-uniform -> s_load
  float*       ob = out + ((size_t)b * CHN + c0) * plane + (size_t)row * HWDIM + col;

  for (int c = 0; c < CPB; ++c) {
    const float* xc = xb + (size_t)c * plane;

    // Batch all 36 gathers first: issued as one load burst (single deep
    // s_wait_loadcnt instead of a wait per tap), maximizing memory-level
    // parallelism per wave. Addresses are loop-invariant; only saddr moves.
    float v[KK][4];
#pragma unroll
    for (int k = 0; k < KK; ++k) {
#pragma unroll
      for (int j = 0; j < 4; ++j) {
        v[k][j] = ldg_boff(xc, boff[k][j]);
      }
    }

    float acc = 0.0f;
#pragma unroll
    for (int k = 0; k < KK; ++k) {
      float s = coef[k][0] * v[k][0];
      s = fmaf(coef[k][1], v[k][1], s);
      s = fmaf(coef[k][2], v[k][2], s);
      s = fmaf(coef[k][3], v[k][3], s);
      acc = fmaf(wb[c * KK + k], s, acc);
    }
    ob[(size_t)c * plane] = acc;   // coalesced b32 store
  }
}

extern "C" void kernel_launch(void* const* d_in, const int* in_sizes, int n_in,
                              void* d_out, int out_size, void* d_ws, size_t ws_size,
                              hipStream_t stream) {
  const float* x   = (const float*)d_in[0];
  const float* off = (const float*)d_in[1];
  const float* msk = (const float*)d_in[2];
  const float* wgt = (const float*)d_in[3];
  float* out = (float*)d_out;

  dim3 grid(BATCH, HWDIM / ROWS, CSPLIT);   // 8 x 16 x 4 = 512 workgroups
  dim3 block(TPX);                          // 256 threads = 8 wave32
  hipLaunchKernelGGL(ModulatedDeformConv_10780367912993_kernel,
                     grid, block, 0, stream, x, off, msk, wgt, out);
}